// AGNNet_8624294330971
// MI455X (gfx1250) — compile-verified
//
#include <hip/hip_runtime.h>
#include <math.h>

#define N_NODES 50000
#define N_EDGES 1600000
#define F_IN    2000
#define NH      16
#define NC      20
#define N_TOT   (N_EDGES + N_NODES)     // edges + self-loops

typedef float v2f __attribute__((ext_vector_type(2)));
typedef float v8f __attribute__((ext_vector_type(8)));

// ---------------------------------------------------------------------------
// Kernel 1: h = relu(x @ W1 + b1) via V_WMMA_F32_16X16X4_F32
// One wave (32 lanes) computes one 16x16 output tile (16 node rows x NH).
// W1 is staged through LDS in 32KB chunks shared by all 8 waves of the block.
// ---------------------------------------------------------------------------
#define KCHUNK 500                      // W1 rows per LDS tile: 500*16*4B = 32KB

__global__ __launch_bounds__(256)
void gemm_relu_wmma(const float* __restrict__ x, const float* __restrict__ W1,
                    const float* __restrict__ b1, float* __restrict__ h)
{
    __shared__ float w1s[KCHUNK * NH];
    const int tid   = threadIdx.x;
    const int wave  = tid >> 5;
    const int lane  = tid & 31;
    const int tile  = blockIdx.x * 8 + wave;         // 16-row tile index
    const bool active = (tile * 16) < N_NODES;       // N_NODES % 16 == 0

    const int row   = tile * 16 + (lane & 15);       // A: M = lane%16
    const int col   = lane & 15;                     // B/C: N = lane%16
    const int khalf = (lane >> 4) * 2;               // 0 for lanes 0-15, 2 for 16-31

    v8f c = {};
    for (int kb = 0; kb < F_IN; kb += KCHUNK) {
        // cooperative load of a W1 chunk into LDS (coalesced)
        for (int i = tid; i < KCHUNK * NH; i += 256)
            w1s[i] = W1[kb * NH + i];
        __syncthreads();

        if (active) {
            const float* xrow = x + (size_t)row * F_IN + kb + khalf;
            for (int k2 = 0; k2 < KCHUNK; k2 += 4) {
                // A tile: 16x4 f32, lane holds {K=khalf, K=khalf+1} of its row
                v2f a = __builtin_nontemporal_load((const v2f*)(xrow + k2));
                // B tile: 4x16 f32 from LDS, lane holds column `col`
                v2f b;
                b.x = w1s[(k2 + khalf    ) * NH + col];
                b.y = w1s[(k2 + khalf + 1) * NH + col];
                c = __builtin_amdgcn_wmma_f32_16x16x4_f32(
                        false, a, false, b, (short)0, c, false, false);
            }
        }
        __syncthreads();
    }

    if (active) {
        const float bias = b1[col];
        const int rbase = tile * 16 + 8 * (lane >> 4);   // C: M = j + 8*(lane/16)
        for (int j = 0; j < 8; ++j) {
            float v = c[j] + bias;
            h[(size_t)(rbase + j) * NH + col] = v > 0.f ? v : 0.f;
        }
    }
}

// ---------------------------------------------------------------------------
// Per-node: hn = h/(|h|+1e-12); init scatter-softmax state and agg output.
// ---------------------------------------------------------------------------
__global__ __launch_bounds__(256)
void node_norm_init(const float* __restrict__ hin, float* __restrict__ hn,
                    unsigned* __restrict__ mOrd, float* __restrict__ ssum,
                    float* __restrict__ agg)
{
    int i = blockIdx.x * blockDim.x + threadIdx.x;
    if (i >= N_NODES) return;
    const float4* r = (const float4*)(hin + (size_t)i * NH);
    float4 v0 = r[0], v1 = r[1], v2 = r[2], v3 = r[3];
    float sq = v0.x*v0.x + v0.y*v0.y + v0.z*v0.z + v0.w*v0.w
             + v1.x*v1.x + v1.y*v1.y + v1.z*v1.z + v1.w*v1.w
             + v2.x*v2.x + v2.y*v2.y + v2.z*v2.z + v2.w*v2.w
             + v3.x*v3.x + v3.y*v3.y + v3.z*v3.z + v3.w*v3.w;
    float inv = 1.f / (sqrtf(sq) + 1e-12f);
    float4* w = (float4*)(hn + (size_t)i * NH);
    float4 o0 = {v0.x*inv, v0.y*inv, v0.z*inv, v0.w*inv};
    float4 o1 = {v1.x*inv, v1.y*inv, v1.z*inv, v1.w*inv};
    float4 o2 = {v2.x*inv, v2.y*inv, v2.z*inv, v2.w*inv};
    float4 o3 = {v3.x*inv, v3.y*inv, v3.z*inv, v3.w*inv};
    w[0] = o0; w[1] = o1; w[2] = o2; w[3] = o3;
    mOrd[i] = 0u;            // minimal ordered key (decodes to NaN -> 0 later)
    ssum[i] = 0.f;
    float4 z = {0.f, 0.f, 0.f, 0.f};
    float4* a = (float4*)(agg + (size_t)i * NH);
    a[0] = z; a[1] = z; a[2] = z; a[3] = z;
}

// ordered-int key for float atomic max (total order incl. negatives)
__device__ __forceinline__ unsigned fkey(float f) {
    unsigned u = __float_as_uint(f);
    return (u & 0x80000000u) ? ~u : (u | 0x80000000u);
}

// ---------------------------------------------------------------------------
// Edge pass 1: alpha_e = beta * <hn[src], hn[dst]>; atomic max per dst.
// Wave-cooperative: 16 lanes per edge (2 edges per wave32). Each half-wave
// reads one contiguous 64B node row per operand (coalesced), then reduces
// the dot product with a 4-step shfl_xor butterfly inside the half.
// Edges [E, E+N) are the implicit self-loops.
// ---------------------------------------------------------------------------
__global__ __launch_bounds__(256)
void edge_alpha(const float* __restrict__ hn, const int* __restrict__ src,
                const int* __restrict__ dst, const float* __restrict__ betaPtr,
                float* __restrict__ alpha, unsigned* __restrict__ mOrd)
{
    const int lane = threadIdx.x & 31;
    const int f    = lane & 15;
    const long wid = (long)blockIdx.x * 8 + (threadIdx.x >> 5);
    const long e   = wid * 2 + (lane >> 4);          // edge handled by this half
    const bool valid = e < (long)N_TOT;

    int s = 0, d = 0;
    if (valid) {
        if (e < N_EDGES) { s = src[e]; d = dst[e]; }
        else             { s = d = (int)(e - N_EDGES); }
    }
    float v = valid ? hn[(size_t)s * NH + f] * hn[(size_t)d * NH + f] : 0.f;
    // butterfly sum within each 16-lane half (xor keeps lanes inside the half)
    v += __shfl_xor(v, 8);
    v += __shfl_xor(v, 4);
    v += __shfl_xor(v, 2);
    v += __shfl_xor(v, 1);

    if (valid && f == 0) {
        float beta = betaPtr ? betaPtr[0] : 1.0f;
        float al = beta * v;
        alpha[e] = al;
        atomicMax(&mOrd[d], fkey(al));
    }
}

// decode ordered key back to float; non-finite -> 0 (matches reference)
__global__ __launch_bounds__(256)
void node_decode_m(const unsigned* __restrict__ mOrd, float* __restrict__ mF)
{
    int i = blockIdx.x * blockDim.x + threadIdx.x;
    if (i >= N_NODES) return;
    unsigned k = mOrd[i];
    unsigned u = (k & 0x80000000u) ? (k & 0x7FFFFFFFu) : ~k;
    float f = __uint_as_float(u);
    if (!isfinite(f)) f = 0.f;
    mF[i] = f;
}

// ---------------------------------------------------------------------------
// Edge pass 2: e = exp(alpha - m[dst]); segment-sum into ssum[dst].
// ---------------------------------------------------------------------------
__global__ __launch_bounds__(256)
void edge_exp_sum(float* __restrict__ alpha, const int* __restrict__ dst,
                  const float* __restrict__ mF, float* __restrict__ ssum)
{
    int e = blockIdx.x * blockDim.x + threadIdx.x;
    if (e >= N_TOT) return;
    int d = (e < N_EDGES) ? dst[e] : (e - N_EDGES);
    float ev = __expf(alpha[e] - mF[d]);
    alpha[e] = ev;
    atomicAdd(&ssum[d], ev);
}

// ---------------------------------------------------------------------------
// Edge pass 3: out[dst] += (e/(s[dst]+1e-16)) * h[src]
// Wave-cooperative: 16 lanes per edge -> each atomic instruction covers two
// contiguous 64B node rows instead of 32 scattered addresses.
// ---------------------------------------------------------------------------
__global__ __launch_bounds__(256)
void edge_aggregate(const float* __restrict__ alpha, const int* __restrict__ src,
                    const int* __restrict__ dst, const float* __restrict__ ssum,
                    const float* __restrict__ hin, float* __restrict__ agg)
{
    const int lane = threadIdx.x & 31;
    const int f    = lane & 15;
    const long wid = (long)blockIdx.x * 8 + (threadIdx.x >> 5);
    const long e   = wid * 2 + (lane >> 4);
    if (e >= (long)N_TOT) return;
    int s, d;
    if (e < N_EDGES) { s = src[e]; d = dst[e]; }
    else             { s = d = (int)(e - N_EDGES); }
    float coef = alpha[e] / (ssum[d] + 1e-16f);      // uniform within the half
    atomicAdd(&agg[(size_t)d * NH + f], coef * hin[(size_t)s * NH + f]);
}

// ---------------------------------------------------------------------------
// Head: logits = h @ W4 + b4; log_softmax over 20 classes.
// ---------------------------------------------------------------------------
__global__ __launch_bounds__(256)
void head_logsoftmax(const float* __restrict__ hin, const float* __restrict__ W4,
                     const float* __restrict__ b4, float* __restrict__ out)
{
    int i = blockIdx.x * blockDim.x + threadIdx.x;
    if (i >= N_NODES) return;
    float hv[NH];
    #pragma unroll
    for (int f = 0; f < NH; ++f) hv[f] = hin[(size_t)i * NH + f];
    float lg[NC];
    float mx = -INFINITY;
    #pragma unroll
    for (int c = 0; c < NC; ++c) {
        float acc = b4[c];
        #pragma unroll
        for (int f = 0; f < NH; ++f) acc += hv[f] * W4[f * NC + c];
        lg[c] = acc;
        mx = fmaxf(mx, acc);
    }
    float sum = 0.f;
    #pragma unroll
    for (int c = 0; c < NC; ++c) sum += __expf(lg[c] - mx);
    float lse = mx + __logf(sum);
    #pragma unroll
    for (int c = 0; c < NC; ++c) out[(size_t)i * NC + c] = lg[c] - lse;
}

// ---------------------------------------------------------------------------
extern "C" void kernel_launch(void* const* d_in, const int* in_sizes, int n_in,
                              void* d_out, int out_size, void* d_ws, size_t ws_size,
                              hipStream_t stream)
{
    const float* x     = (const float*)d_in[0];
    const int*   ei    = (const int*)  d_in[1];
    const float* W1    = (const float*)d_in[2];
    const float* b1    = (const float*)d_in[3];
    const float* W4    = (const float*)d_in[4];
    const float* b4    = (const float*)d_in[5];
    const float* beta3 = (const float*)d_in[6];
    const int* srcIdx = ei;
    const int* dstIdx = ei + N_EDGES;

    // workspace layout (all offsets multiple of 256B)
    char* ws = (char*)d_ws;
    const size_t SZH = (size_t)N_NODES * NH * sizeof(float);      // 3,200,000
    const size_t SZA = ((size_t)N_TOT * 4 + 255) & ~(size_t)255;
    const size_t SZN = ((size_t)N_NODES * 4 + 255) & ~(size_t)255;
    float*    h1    = (float*)   (ws);
    float*    h2    = (float*)   (ws + SZH);
    float*    h3    = (float*)   (ws + 2*SZH);
    float*    hn    = (float*)   (ws + 3*SZH);
    float*    alpha = (float*)   (ws + 4*SZH);
    unsigned* mOrd  = (unsigned*)(ws + 4*SZH + SZA);
    float*    mF    = (float*)   (ws + 4*SZH + SZA + SZN);
    float*    ssum  = (float*)   (ws + 4*SZH + SZA + 2*SZN);

    const int NODE_BLKS  = (N_NODES + 255) / 256;
    const int EDGE_BLKS  = (N_TOT + 255) / 256;     // 1 thread / edge
    const int EDGE2_BLKS = (N_TOT + 15) / 16;       // 16 lanes / edge, 16 edges/block
    const int GEMM_BLKS  = (N_NODES / 16 + 7) / 8;  // 8 waves/block

    // linear1 + relu
    gemm_relu_wmma<<<GEMM_BLKS, 256, 0, stream>>>(x, W1, b1, h1);

    // conv2 (beta = 1.0)
    node_norm_init<<<NODE_BLKS, 256, 0, stream>>>(h1, hn, mOrd, ssum, h2);
    edge_alpha    <<<EDGE2_BLKS, 256, 0, stream>>>(hn, srcIdx, dstIdx, nullptr, alpha, mOrd);
    node_decode_m <<<NODE_BLKS, 256, 0, stream>>>(mOrd, mF);
    edge_exp_sum  <<<EDGE_BLKS, 256, 0, stream>>>(alpha, dstIdx, mF, ssum);
    edge_aggregate<<<EDGE2_BLKS, 256, 0, stream>>>(alpha, srcIdx, dstIdx, ssum, h1, h2);

    // conv3 (learnable beta)
    node_norm_init<<<NODE_BLKS, 256, 0, stream>>>(h2, hn, mOrd, ssum, h3);
    edge_alpha    <<<EDGE2_BLKS, 256, 0, stream>>>(hn, srcIdx, dstIdx, beta3, alpha, mOrd);
    node_decode_m <<<NODE_BLKS, 256, 0, stream>>>(mOrd, mF);
    edge_exp_sum  <<<EDGE_BLKS, 256, 0, stream>>>(alpha, dstIdx, mF, ssum);
    edge_aggregate<<<EDGE2_BLKS, 256, 0, stream>>>(alpha, srcIdx, dstIdx, ssum, h2, h3);

    // linear4 + log_softmax
    head_logsoftmax<<<NODE_BLKS, 256, 0, stream>>>(h3, W4, b4, (float*)d_out);
}